// PerformanceModel_6064493822065
// MI455X (gfx1250) — compile-verified
//
#include <hip/hip_runtime.h>
#include <cstdint>

// ---------------------------------------------------------------------------
// perf = piecewise(sig(min(lb1,ub1)), sig(max(lb1,ub1)), bins[idx0])
//      * piecewise(sig(min(lb2,ub2)), sig(max(lb2,ub2)), bins[idx1])
//      * sig(resp)
// Memory-bound gather/stream kernel: 64 MB traffic, ~2.9 us at 23.3 TB/s.
// CDNA5 paths: global_load_async_to_lds_b32 + s_wait_asynccnt for the 4KB
// bin-table stage, global_prefetch_b8 (WGP scope) on the index stream,
// non-temporal b128 loads/stores for the 16B/row stream, ds_load gathers.
// ---------------------------------------------------------------------------

#define TPB   256      // 8 wave32s per block
#define RPT   4        // rows per thread -> 3x b128 index loads, 1x b128 store
#define NBINS 1000
#define PEPS  1e-4f

typedef int   v4i __attribute__((ext_vector_type(4)));
typedef float v4f __attribute__((ext_vector_type(4)));

typedef __attribute__((address_space(1))) int gint_t;
typedef __attribute__((address_space(3))) int lint_t;

// CDNA5 async global->LDS path (gfx1250). __has_builtin is false during the
// host pass and on toolchains lacking the builtin, so this cannot break compile.
#if defined(__HIP_DEVICE_COMPILE__)
#if __has_builtin(__builtin_amdgcn_global_load_async_to_lds_b32) && \
    __has_builtin(__builtin_amdgcn_s_wait_asynccnt)
#define USE_ASYNC_LDS 1
#endif
#endif

__device__ __forceinline__ float rcp_fast(float x) {
#ifdef __HIP_DEVICE_COMPILE__
    return __builtin_amdgcn_rcpf(x);   // v_rcp_f32
#else
    return 1.0f / x;
#endif
}

__device__ __forceinline__ float fast_sigmoid(float x) {
    return rcp_fast(1.0f + __expf(-x));
}

__global__ void __launch_bounds__(TPB) perf_gather_kernel(
    const float* __restrict__ bin_centers,
    const int*   __restrict__ obs_idx,
    const float* __restrict__ plb1, const float* __restrict__ pub1,
    const float* __restrict__ plb2, const float* __restrict__ pub2,
    const float* __restrict__ presp,
    float* __restrict__ out, int n)
{
    __shared__ float s_bins[NBINS];
    const int tid = threadIdx.x;
    const int r0  = (blockIdx.x * TPB + tid) * RPT;

    // Kick off the streaming-index prefetch before we stall on LDS staging.
    // locality=3 -> WGP-scope global_prefetch_b8 (pull into all cache levels),
    // so the b128 index loads below hit near caches after the barrier.
    if (r0 < n) __builtin_prefetch(obs_idx + (size_t)r0 * 3, 0, 3);

    // ---- stage the 4KB bin table into LDS --------------------------------
#ifdef USE_ASYNC_LDS
    for (int i = tid; i < NBINS; i += TPB) {
        __builtin_amdgcn_global_load_async_to_lds_b32(
            (gint_t*)(uintptr_t)(bin_centers + i),
            (lint_t*)(uint32_t)(uintptr_t)(&s_bins[i]),
            /*offset=*/0, /*cpol=*/0);
    }
    __builtin_amdgcn_s_wait_asynccnt(0);
#else
    for (int i = tid; i < NBINS; i += TPB) s_bins[i] = bin_centers[i];
#endif

    // ---- uniform scalar parameters (tiny cached loads + a few trans ops) --
    const float l1 = plb1[0], u1 = pub1[0];
    const float l2 = plb2[0], u2 = pub2[0];
    const float nlb1 = fast_sigmoid(fminf(l1, u1));
    const float nub1 = fast_sigmoid(fmaxf(l1, u1));
    const float nlb2 = fast_sigmoid(fminf(l2, u2));
    const float nub2 = fast_sigmoid(fmaxf(l2, u2));
    const float rr   = fast_sigmoid(presp[0]);
    const float inv1 = rcp_fast(nub1 - nlb1 + PEPS);
    const float inv2 = rcp_fast(nub2 - nlb2 + PEPS);

    __syncthreads();

    auto pwf = [&](float lb, float ub, float inv, float x) -> float {
        float ramp = (ub - x) * inv;          // == (ub-x)/(ub-lb+EPS)
        float v = (x > ub) ? 0.0f : ramp;
        return (x <= lb) ? 1.0f : v;
    };
    auto rowp = [&](int i0, int i1) -> float {
        return pwf(nlb1, nub1, inv1, s_bins[i0]) *
               pwf(nlb2, nub2, inv2, s_bins[i1]) * rr;
    };

    if (r0 + RPT <= n) {
        // 4 rows = 12 ints = 3 x 16B aligned non-temporal b128 loads
        const v4i* p = (const v4i*)(obs_idx + (size_t)r0 * 3);
        v4i a = __builtin_nontemporal_load(p);
        v4i b = __builtin_nontemporal_load(p + 1);
        v4i c = __builtin_nontemporal_load(p + 2);
        v4f o;
        o.x = rowp(a.x, a.y);   // row r0  : cols 0,1
        o.y = rowp(a.w, b.x);   // row r0+1: cols 0,1
        o.z = rowp(b.z, b.w);   // row r0+2: cols 0,1
        o.w = rowp(c.y, c.z);   // row r0+3: cols 0,1
        __builtin_nontemporal_store(o, (v4f*)(out + r0));
    } else {
        for (int r = r0; r < n && r < r0 + RPT; ++r) {
            int i0 = obs_idx[(size_t)r * 3 + 0];
            int i1 = obs_idx[(size_t)r * 3 + 1];
            out[r] = rowp(i0, i1);
        }
    }
}

extern "C" void kernel_launch(void* const* d_in, const int* in_sizes, int n_in,
                              void* d_out, int out_size, void* d_ws, size_t ws_size,
                              hipStream_t stream) {
    (void)in_sizes; (void)n_in; (void)d_ws; (void)ws_size;
    const float* bin_centers = (const float*)d_in[0];
    const int*   obs_idx     = (const int*)d_in[1];
    const float* lb1  = (const float*)d_in[2];
    const float* ub1  = (const float*)d_in[3];
    const float* lb2  = (const float*)d_in[4];
    const float* ub2  = (const float*)d_in[5];
    const float* resp = (const float*)d_in[6];
    float* out = (float*)d_out;

    const int n = out_size;                       // 4,194,304 rows
    const int rows_per_block = TPB * RPT;         // 1024
    int blocks = (n + rows_per_block - 1) / rows_per_block;   // 4096
    if (blocks < 1) blocks = 1;

    perf_gather_kernel<<<blocks, TPB, 0, stream>>>(
        bin_centers, obs_idx, lb1, ub1, lb2, ub2, resp, out, n);
}